// blockDecLayer_48653389529230
// MI455X (gfx1250) — compile-verified
//
#include <hip/hip_runtime.h>

typedef float v2f __attribute__((ext_vector_type(2)));
typedef float v8f __attribute__((ext_vector_type(8)));

#define KTOT    49152      // 3 * 16384
#define BROWS   4096
#define KSEG_WAVES 64
#define KSEG_LEN   (KTOT / KSEG_WAVES)   // 768 K-values per wave
#define ROWTILES   (BROWS / 16)          // 256

__device__ __forceinline__ float softplus_f(float x) {
    // numerically stable softplus
    return fmaxf(x, 0.0f) + log1pf(__expf(-fabsf(x)));
}

// Build the B-operand image for V_WMMA_F32_16X16X4_F32.
// Layout: wB2[p][j][e], p = k/2, e = k&1  (float, [KTOT/2][16][2] = 3 MiB)
//   wB2[p][j][e] = W(2p+e, j),  W(k, j) = (j < 3) ? (M^T M)_{block k/3}[j][k%3] : 0
// This matches the assumed hardware B layout for 16x16x4 f32:
//   VGPR0: lanes 0-15 -> B[K=0, N=lane], lanes 16-31 -> B[K=2, N=lane-16]
//   VGPR1: lanes 0-15 -> B[K=1, N=lane], lanes 16-31 -> B[K=3, N=lane-16]
__global__ void build_w_image(const float* __restrict__ w, float* __restrict__ wB2) {
    int k = blockIdx.x * blockDim.x + threadIdx.x;
    if (k >= KTOT) return;
    int n = k / 3;
    int i = k - 3 * n;
    const float* wr = w + n * 6;
    float a = softplus_f(wr[0]);
    float b = wr[1];
    float c = wr[2];
    float d = softplus_f(wr[3]);
    float e = wr[4];
    float f = softplus_f(wr[5]);
    // column i of symmetric M^T M
    float c0, c1, c2;
    if (i == 0)      { c0 = a * a;         c1 = a * b;             c2 = a * c; }
    else if (i == 1) { c0 = a * b;         c1 = b * b + d * d;     c2 = b * c + d * e; }
    else             { c0 = a * c;         c1 = b * c + d * e;     c2 = c * c + e * e + f * f; }
    float* dst = wB2 + (size_t)(k >> 1) * 32 + (k & 1);
    dst[0] = c0;
    dst[2] = c1;
    dst[4] = c2;
    #pragma unroll
    for (int j = 3; j < 16; ++j) dst[2 * j] = 0.0f;   // unused N columns
}

// Streaming GEMM: out[b, 0:3] += x[b, :] * W[:, 0:3] via f32 WMMA, split-K.
__global__ void __launch_bounds__(256) blockdec_wmma(const float* __restrict__ x,
                                                     const float* __restrict__ wB2,
                                                     float* __restrict__ out) {
    const int lane  = threadIdx.x & 31;
    const int wid   = blockIdx.x * 8 + (threadIdx.x >> 5);
    const int rowtile = wid >> 6;                 // / KSEG_WAVES
    const int kseg    = wid & (KSEG_WAVES - 1);
    const int ks      = kseg * KSEG_LEN;

    const int row  = lane & 15;                   // M within 16-row tile
    const int koff = (lane >> 4) << 1;            // 0 (lanes 0-15) or 2 (lanes 16-31)

    // A operand: each lane streams its row with contiguous 8B loads
    const float* ap = x + (size_t)(rowtile * 16 + row) * KTOT + ks + koff;
    // B operand: precomputed register image, contiguous 8B per lane (L2-resident)
    const float* bp = wB2 + (size_t)((ks + koff) >> 1) * 32 + (size_t)(lane & 15) * 2;

    v8f acc = {};
    #pragma unroll 4
    for (int it = 0; it < KSEG_LEN / 4; ++it) {
        v2f a = *(const v2f*)ap;                  // x[row, k0+koff .. +1]
        v2f b = *(const v2f*)bp;                  // W(k0+koff, n), W(k0+koff+1, n)
        acc = __builtin_amdgcn_wmma_f32_16x16x4_f32(
            /*neg_a=*/false, a, /*neg_b=*/false, b,
            /*c_mod=*/(short)0, acc, /*reuse_a=*/false, /*reuse_b=*/false);
        ap += 4;        // K advances by 4 per WMMA
        bp += 64;       // 2 pairs * 32 floats
    }

    // C layout: VGPR r -> (M=r, N=lane) for lanes 0-15, (M=r+8, N=lane-16) for 16-31
    const int Nn = lane & 15;
    if (Nn < 3) {
        const int mbase = rowtile * 16 + ((lane >> 4) << 3);
        #pragma unroll
        for (int r = 0; r < 8; ++r) {
            atomicAdd(out + (size_t)(mbase + r) * 3 + Nn, acc[r]);
        }
    }
}

extern "C" void kernel_launch(void* const* d_in, const int* in_sizes, int n_in,
                              void* d_out, int out_size, void* d_ws, size_t ws_size,
                              hipStream_t stream) {
    (void)in_sizes; (void)n_in; (void)ws_size;
    const float* x = (const float*)d_in[0];      // [4096, 49152] f32
    const float* w = (const float*)d_in[1];      // [16384, 6]    f32
    float* out = (float*)d_out;                  // [4096, 3]     f32
    float* wB2 = (float*)d_ws;                   // 3 MiB scratch: B-operand image

    hipMemsetAsync(d_out, 0, (size_t)out_size * sizeof(float), stream);
    build_w_image<<<(KTOT + 255) / 256, 256, 0, stream>>>(w, wB2);
    blockdec_wmma<<<(ROWTILES * KSEG_WAVES) / 8, 256, 0, stream>>>(x, wB2, out);
}